// FullTreeLSTM_26087631356090
// MI455X (gfx1250) — compile-verified
//
#include <hip/hip_runtime.h>

// ---------------------------------------------------------------------------
// Tree-LSTM forward (depth 9, arity 4, H=256, D=300, E=37) for gfx1250.
// fp32 end-to-end; dense GEMMs via V_WMMA_F32_16X16X4_F32 (wave32).
// ---------------------------------------------------------------------------

#define HDIM 256
#define DDIM 300

typedef float v2f __attribute__((ext_vector_type(2)));
typedef float v8f __attribute__((ext_vector_type(8)));

// D(16x16) = A(16x4) * B(4x16) + C   (f32 WMMA, wave32)
// A layout: lane L -> m = L&15, khalf = L>>4; vgpr0 = A[m][2*khalf], vgpr1 = A[m][2*khalf+1]
// B layout: lane L -> n = L&15;              vgpr0 = B[2*khalf][n],  vgpr1 = B[2*khalf+1][n]
// C/D     : vgpr v, lanes 0-15 -> row v, lanes 16-31 -> row v+8; col = L&15
__device__ __forceinline__ v8f wmma_f32_16x16x4(v2f a, v2f b, v8f c) {
  return __builtin_amdgcn_wmma_f32_16x16x4_f32(false, a, false, b, (short)0, c,
                                               false, false);
}

__device__ __forceinline__ float sigmoidf_(float xv) {
  return 1.0f / (1.0f + __expf(-xv));
}

// ---------------------------------------------------------------------------
// Leaf kernel: fused (x @ ioux_w.T + ioux_b + iouh_b) GEMM + LSTM activations.
// One 16-node tile per block; 4 waves; wave w owns column-triples so each
// lane holds matching (i, o, u) elements and can finish in-register.
// ---------------------------------------------------------------------------
__global__ __launch_bounds__(128) void tlstm_leaf_kernel(
    const float* __restrict__ x, const float* __restrict__ ioux_w,
    const float* __restrict__ ioux_b, const float* __restrict__ iouh_b,
    float* __restrict__ h_all, float* __restrict__ c_all) {
  const int tid = threadIdx.x;
  const int wave = tid >> 5;
  const int lane = tid & 31;
  const int kh = lane >> 4;
  const int n16 = lane & 15;
  const int base = blockIdx.x << 4;

  v8f acc[12] = {};  // [t/3 = triple, t%3 = i/o/u part]
  const float* xrow = x + (size_t)(base + n16) * DDIM;

  const float* wp[12];
#pragma unroll
  for (int t = 0; t < 12; ++t) {
    const int jb = wave * 4 + t / 3 + (t % 3) * 16;  // 0..47 column blocks
    wp[t] = ioux_w + (size_t)(jb * 16 + n16) * DDIM;
  }

  for (int k0 = 0; k0 < DDIM; k0 += 4) {
    v2f a;
    a.x = xrow[k0 + 2 * kh];
    a.y = xrow[k0 + 2 * kh + 1];
#pragma unroll
    for (int t = 0; t < 12; ++t) {
      v2f b;
      b.x = wp[t][k0 + 2 * kh];
      b.y = wp[t][k0 + 2 * kh + 1];
      acc[t] = wmma_f32_16x16x4(a, b, acc[t]);
    }
  }

#pragma unroll
  for (int t4 = 0; t4 < 4; ++t4) {
    const int b0 = wave * 4 + t4;
    const int ji = b0 * 16 + n16;  // hidden index 0..255
    const float bi = ioux_b[ji] + iouh_b[ji];
    const float bo = ioux_b[ji + 256] + iouh_b[ji + 256];
    const float bu = ioux_b[ji + 512] + iouh_b[ji + 512];
#pragma unroll
    for (int v = 0; v < 8; ++v) {
      const int row = v + 8 * kh;
      const size_t node = (size_t)(base + row);
      const float iv = acc[t4 * 3 + 0][v] + bi;
      const float ov = acc[t4 * 3 + 1][v] + bo;
      const float uv = acc[t4 * 3 + 2][v] + bu;
      const float cv = sigmoidf_(iv) * tanhf(uv);
      const float hv = sigmoidf_(ov) * tanhf(cv);
      c_all[node * HDIM + ji] = cv;
      h_all[node * HDIM + ji] = hv;
    }
  }
}

// ---------------------------------------------------------------------------
// Interior-level kernel: one 16-node tile per 128-thread workgroup.
// LDS (dynamic, 148KB): ch_h[64][256] | xf[16][256] | hsum/fc[16][256] |
//                       iou[16][768]  | idx[64] et[64]
// ---------------------------------------------------------------------------
__global__ __launch_bounds__(128) void tlstm_level_kernel(
    const float* __restrict__ x, const int* __restrict__ children_idx,
    const int* __restrict__ edge_type, const float* __restrict__ e2h_w,
    const float* __restrict__ ioux_w, const float* __restrict__ ioux_b,
    const float* __restrict__ iouh_w, const float* __restrict__ iouh_b,
    const float* __restrict__ fx_w, const float* __restrict__ fx_b,
    const float* __restrict__ fh_w, const float* __restrict__ fh_b,
    float* __restrict__ h_all, float* __restrict__ c_all, int lvl_start,
    int lvl_count) {
  extern __shared__ float smem[];
  float* lds_chh = smem;                   // 64*256 floats
  float* lds_xf = lds_chh + 64 * HDIM;     // 16*256
  float* lds_hsum = lds_xf + 16 * HDIM;    // 16*256 (aliased as fc later)
  float* lds_iou = lds_hsum + 16 * HDIM;   // 16*768
  float* lds_fc = lds_hsum;                // reuse: hsum dead after iou GEMM
  int* lds_idx = (int*)(lds_iou + 16 * 3 * HDIM);  // 64
  int* lds_et = lds_idx + 64;                      // 64

  const int tid = threadIdx.x;
  const int wave = tid >> 5;
  const int lane = tid & 31;
  const int kh = lane >> 4;
  const int n16 = lane & 15;
  const int base = lvl_start + (blockIdx.x << 4);
  const int rem = lvl_start + lvl_count - base;
  const int mtile = rem < 16 ? rem : 16;

  // --- child indices / edge types (padded rows clamp to last valid node) ---
  if (tid < 64) {
    const int mm = tid >> 2, cc = tid & 3;
    const int mcl = mm < mtile ? mm : mtile - 1;
    const int node = base + mcl;
    lds_idx[tid] = children_idx[node * 4 + cc];
    lds_et[tid] = edge_type[node * 4 + cc];
  }
  __syncthreads();

  // --- gather 64 child h rows into LDS (float4 granularity) ---
  {
    const float4* __restrict__ h4 = (const float4*)h_all;
    float4* c4 = (float4*)lds_chh;
    for (int q = tid; q < 64 * 64; q += 128) {
      const int row = q >> 6, col = q & 63;
      c4[q] = h4[(size_t)lds_idx[row] * 64 + col];
    }
  }

  // --- xf = x @ fx_w.T + fx_b  (WMMA, 16 column blocks across 4 waves) ---
  {
    v8f accx[4] = {};
    const int mcl = n16 < mtile ? n16 : mtile - 1;
    const float* xrow = x + (size_t)(base + mcl) * DDIM;
    const float* wpx[4];
#pragma unroll
    for (int t = 0; t < 4; ++t)
      wpx[t] = fx_w + (size_t)((wave * 4 + t) * 16 + n16) * DDIM;
    for (int k0 = 0; k0 < DDIM; k0 += 4) {
      v2f a;
      a.x = xrow[k0 + 2 * kh];
      a.y = xrow[k0 + 2 * kh + 1];
#pragma unroll
      for (int t = 0; t < 4; ++t) {
        v2f b;
        b.x = wpx[t][k0 + 2 * kh];
        b.y = wpx[t][k0 + 2 * kh + 1];
        accx[t] = wmma_f32_16x16x4(a, b, accx[t]);
      }
    }
#pragma unroll
    for (int t = 0; t < 4; ++t) {
      const int j = (wave * 4 + t) * 16 + n16;
      const float bias = fx_b[j];
#pragma unroll
      for (int v = 0; v < 8; ++v)
        lds_xf[(v + 8 * kh) * HDIM + j] = accx[t][v] + bias;
    }
  }
  __syncthreads();

  // --- hsum[m] = sum_c e2h_w[et[m][c]] @ h_child[m][c] -------------------
  // Block-diagonal per-(node,child) operator: weight matrix differs per row,
  // so not expressible as a dense WMMA. ~15% of total FLOPs; VALU matvec
  // with float4 weight loads (weights are L2-resident: 9.7MB << 192MB).
  {
    const int m = tid >> 3;
    const int i0 = (tid & 7) * 32;
    float accd[32];
#pragma unroll
    for (int ii = 0; ii < 32; ++ii) accd[ii] = 0.f;
    for (int c = 0; c < 4; ++c) {
      const int et = lds_et[m * 4 + c];
      const float* __restrict__ wmat = e2h_w + (size_t)et * (HDIM * HDIM);
      const float* __restrict__ hr = lds_chh + (m * 4 + c) * HDIM;
      __builtin_prefetch(wmat + i0 * HDIM, 0, 0);
      for (int j0 = 0; j0 < HDIM; j0 += 8) {
        const float h0 = hr[j0 + 0], h1 = hr[j0 + 1];
        const float h2 = hr[j0 + 2], h3 = hr[j0 + 3];
        const float h4v = hr[j0 + 4], h5 = hr[j0 + 5];
        const float h6 = hr[j0 + 6], h7 = hr[j0 + 7];
        const float* wb = wmat + i0 * HDIM + j0;
#pragma unroll
        for (int ii = 0; ii < 32; ++ii) {
          const float4 w0 = *(const float4*)(wb + ii * HDIM);
          const float4 w1 = *(const float4*)(wb + ii * HDIM + 4);
          accd[ii] += w0.x * h0 + w0.y * h1 + w0.z * h2 + w0.w * h3 +
                      w1.x * h4v + w1.y * h5 + w1.z * h6 + w1.w * h7;
        }
      }
    }
#pragma unroll
    for (int ii = 0; ii < 32; ++ii) lds_hsum[m * HDIM + i0 + ii] = accd[ii];
  }
  __syncthreads();

  // --- iou = hsum @ iouh_w.T + x @ ioux_w.T + biases (fused double-K WMMA) -
  {
    v8f acc[12] = {};
    int jcol[12];
#pragma unroll
    for (int t = 0; t < 12; ++t)
      jcol[t] = (wave * 4 + t / 3) * 16 + (t % 3) * 256 + n16;

    const float* hsrow = lds_hsum + n16 * HDIM;
    for (int k0 = 0; k0 < HDIM; k0 += 4) {
      v2f a;
      a.x = hsrow[k0 + 2 * kh];
      a.y = hsrow[k0 + 2 * kh + 1];
#pragma unroll
      for (int t = 0; t < 12; ++t) {
        const float* wr = iouh_w + (size_t)jcol[t] * HDIM;
        v2f b;
        b.x = wr[k0 + 2 * kh];
        b.y = wr[k0 + 2 * kh + 1];
        acc[t] = wmma_f32_16x16x4(a, b, acc[t]);
      }
    }
    const int mcl = n16 < mtile ? n16 : mtile - 1;
    const float* xrow = x + (size_t)(base + mcl) * DDIM;
    for (int k0 = 0; k0 < DDIM; k0 += 4) {
      v2f a;
      a.x = xrow[k0 + 2 * kh];
      a.y = xrow[k0 + 2 * kh + 1];
#pragma unroll
      for (int t = 0; t < 12; ++t) {
        const float* wr = ioux_w + (size_t)jcol[t] * DDIM;
        v2f b;
        b.x = wr[k0 + 2 * kh];
        b.y = wr[k0 + 2 * kh + 1];
        acc[t] = wmma_f32_16x16x4(a, b, acc[t]);
      }
    }
#pragma unroll
    for (int t = 0; t < 12; ++t) {
      const int j = jcol[t];
      const float bias = ioux_b[j] + iouh_b[j];
#pragma unroll
      for (int v = 0; v < 8; ++v)
        lds_iou[(v + 8 * kh) * (3 * HDIM) + j] = acc[t][v] + bias;
    }
  }
  __syncthreads();

  // --- f = sigmoid(ch_h @ fh_w.T + fh_b + xf[m]); fc = sum_c f*ch_c -------
  // 64 pairs x 256: wave w owns pair-rows 16w..16w+15 (nodes 4w..4w+3).
  {
    v8f accf[16] = {};
    const float* arow = lds_chh + (wave * 16 + n16) * HDIM;
    for (int k0 = 0; k0 < HDIM; k0 += 4) {
      v2f a;
      a.x = arow[k0 + 2 * kh];
      a.y = arow[k0 + 2 * kh + 1];
#pragma unroll
      for (int jb = 0; jb < 16; ++jb) {
        const float* wr = fh_w + (size_t)(jb * 16 + n16) * HDIM;
        v2f b;
        b.x = wr[k0 + 2 * kh];
        b.y = wr[k0 + 2 * kh + 1];
        accf[jb] = wmma_f32_16x16x4(a, b, accf[jb]);
      }
    }
#pragma unroll
    for (int jb = 0; jb < 16; ++jb) {
      const int j = jb * 16 + n16;
      const float fb = fh_b[j];
      float fc0 = 0.f, fc1 = 0.f;
#pragma unroll
      for (int v = 0; v < 8; ++v) {
        const int p = v + 8 * kh;         // pair row within this wave's tile
        const int pair = wave * 16 + p;   // 0..63
        const int mloc = pair >> 2;       // node within tile
        const float fpre = accf[jb][v] + fb + lds_xf[mloc * HDIM + j];
        const float fv = sigmoidf_(fpre);
        const float chc = c_all[(size_t)lds_idx[pair] * HDIM + j];
        const float contrib = fv * chc;
        if (v < 4) fc0 += contrib; else fc1 += contrib;
      }
      const int m0 = wave * 4 + 2 * kh;   // lane covers nodes m0 and m0+1
      lds_fc[m0 * HDIM + j] = fc0;
      lds_fc[(m0 + 1) * HDIM + j] = fc1;
    }
  }
  __syncthreads();

  // --- final gates + store ------------------------------------------------
  {
    const int m = tid >> 3;
    const int j0 = (tid & 7) * 32;
    if (m < mtile) {
      const size_t node = (size_t)(base + m);
      for (int j = j0; j < j0 + 32; ++j) {
        const float iv = lds_iou[m * (3 * HDIM) + j];
        const float ov = lds_iou[m * (3 * HDIM) + HDIM + j];
        const float uv = lds_iou[m * (3 * HDIM) + 2 * HDIM + j];
        const float cv = sigmoidf_(iv) * tanhf(uv) + lds_fc[m * HDIM + j];
        const float hv = sigmoidf_(ov) * tanhf(cv);
        c_all[node * HDIM + j] = cv;
        h_all[node * HDIM + j] = hv;
      }
    }
  }
}

__global__ void tlstm_root_kernel(const float* __restrict__ c_all,
                                  const float* __restrict__ h_all,
                                  float* __restrict__ out, int root) {
  const int t = threadIdx.x;
  if (t < 256)
    out[t] = c_all[(size_t)root * HDIM + t];
  else
    out[t] = h_all[(size_t)root * HDIM + (t - 256)];
}

// ---------------------------------------------------------------------------
extern "C" void kernel_launch(void* const* d_in, const int* in_sizes, int n_in,
                              void* d_out, int out_size, void* d_ws,
                              size_t ws_size, hipStream_t stream) {
  (void)n_in; (void)out_size; (void)ws_size;
  const float* x = (const float*)d_in[0];
  const int* children = (const int*)d_in[1];
  const int* etype = (const int*)d_in[2];
  // d_in[3]=depth(9), d_in[4]=arity(4): fixed by problem geometry.
  const float* e2h_w = (const float*)d_in[5];
  const float* ioux_w = (const float*)d_in[6];
  const float* ioux_b = (const float*)d_in[7];
  const float* iouh_w = (const float*)d_in[8];
  const float* iouh_b = (const float*)d_in[9];
  const float* fx_w = (const float*)d_in[10];
  const float* fx_b = (const float*)d_in[11];
  const float* fh_w = (const float*)d_in[12];
  const float* fh_b = (const float*)d_in[13];

  const int N = in_sizes[0] / DDIM;  // 87381
  float* h_all = (float*)d_ws;
  float* c_all = h_all + (size_t)N * HDIM;

  const int nL = 65536;  // 4^(depth-1) leaves
  tlstm_leaf_kernel<<<nL / 16, 128, 0, stream>>>(x, ioux_w, ioux_b, iouh_b,
                                                 h_all, c_all);

  constexpr size_t SMEM =
      (size_t)(64 * HDIM + 16 * HDIM + 16 * HDIM + 16 * 3 * HDIM) *
          sizeof(float) +
      128 * sizeof(int);  // 147968 bytes (< 320KB/WGP; 2 WGs per WGP)
  (void)hipFuncSetAttribute((const void*)tlstm_level_kernel,
                            hipFuncAttributeMaxDynamicSharedMemorySize,
                            (int)SMEM);

  int off = nL;
  for (int l = 1; l < 9; ++l) {
    const int cnt = 65536 >> (2 * l);  // 16384, 4096, ..., 1
    const int tiles = (cnt + 15) / 16;
    tlstm_level_kernel<<<tiles, 128, SMEM, stream>>>(
        x, children, etype, e2h_w, ioux_w, ioux_b, iouh_w, iouh_b, fx_w, fx_b,
        fh_w, fh_b, h_all, c_all, off, cnt);
    off += cnt;
  }

  tlstm_root_kernel<<<1, 512, 0, stream>>>(c_all, h_all, (float*)d_out, N - 1);
}